// GIN_5789615915640
// MI455X (gfx1250) — compile-verified
//
#include <hip/hip_runtime.h>

// GIN forward on MI455X (gfx1250, wave32).
// Phase 1 (once per call): build CSR over dst (count -> scan -> scatter). Int atomics only.
// Phase 2 (per layer):
//   - wave-per-node gather-sum aggregation fused with z = (1+eps)*h + agg*inv_deg  (no f32 atomics)
//   - GEMM z@W+b via V_WMMA_F32_16X16X4_F32 (exact f32), z tile staged to LDS with
//     GLOBAL_LOAD_ASYNC_TO_LDS_B128 when the toolchain exposes it.

typedef __attribute__((ext_vector_type(2))) float v2f;
typedef __attribute__((ext_vector_type(8))) float v8f;
typedef __attribute__((ext_vector_type(4))) int v4i;
typedef __attribute__((address_space(1))) v4i* gv4i_p;  // global int4*
typedef __attribute__((address_space(3))) v4i* lv4i_p;  // LDS int4*

#define FEAT 128
#define ZPAD 4  // LDS row stride 132 floats = 528B: 16B-aligned rows, conflict-free WMMA A reads

#if __has_builtin(__builtin_amdgcn_global_load_async_to_lds_b128)
#define USE_ASYNC_LDS 1
#endif

__global__ __launch_bounds__(256) void gin_zero_i32(int* __restrict__ p, int n) {
  int i = blockIdx.x * 256 + threadIdx.x;
  if (i < n) p[i] = 0;
}

__global__ __launch_bounds__(256) void gin_count_kernel(const int* __restrict__ dst,
                                                        int* __restrict__ cnt, int nEdges) {
  int e = blockIdx.x * 256 + threadIdx.x;
  if (e < nEdges) atomicAdd(cnt + dst[e], 1);
}

// Single-workgroup chunked exclusive scan: rowptr[0..n] from cnt[0..n-1].
__global__ __launch_bounds__(1024) void gin_scan_kernel(const int* __restrict__ cnt,
                                                        int* __restrict__ rowptr, int n) {
  __shared__ int buf[1024];
  __shared__ int carry;
  if (threadIdx.x == 0) carry = 0;
  __syncthreads();
  for (int base = 0; base < n; base += 1024) {
    int i = base + (int)threadIdx.x;
    int v = (i < n) ? cnt[i] : 0;
    buf[threadIdx.x] = v;
    __syncthreads();
    // Hillis-Steele inclusive scan in LDS
    for (int off = 1; off < 1024; off <<= 1) {
      int t = (threadIdx.x >= (unsigned)off) ? buf[threadIdx.x - off] : 0;
      __syncthreads();
      buf[threadIdx.x] += t;
      __syncthreads();
    }
    int incl = buf[threadIdx.x];
    int chunkTotal = buf[1023];
    int baseCarry = carry;
    __syncthreads();  // everyone read carry before thread 0 bumps it
    if (threadIdx.x == 0) carry = baseCarry + chunkTotal;
    if (i < n) rowptr[i] = baseCarry + incl - v;  // exclusive
    __syncthreads();
  }
  if (threadIdx.x == 0) rowptr[n] = carry;
}

__global__ __launch_bounds__(256) void gin_scatter_kernel(const int* __restrict__ src,
                                                          const int* __restrict__ dst,
                                                          const int* __restrict__ rowptr,
                                                          int* __restrict__ fill,
                                                          int* __restrict__ srcIdx, int nEdges) {
  int e = blockIdx.x * 256 + threadIdx.x;
  if (e < nEdges) {
    int d = dst[e];
    int pos = rowptr[d] + atomicAdd(fill + d, 1);
    srcIdx[pos] = src[e];
  }
}

// One wave32 per node: gather-sum neighbor rows (32 lanes x float4 = 128 feats),
// then z = (1+eps)*h[node] + sum * (1/max(deg,1)). No atomics.
__global__ __launch_bounds__(256) void gin_aggz_kernel(const float* __restrict__ h,
                                                       const int* __restrict__ rowptr,
                                                       const int* __restrict__ srcIdx,
                                                       const float* __restrict__ epsArr,
                                                       int layer, int nNodes,
                                                       float* __restrict__ z) {
  int node = (blockIdx.x * 256 + threadIdx.x) >> 5;
  int lane = threadIdx.x & 31;
  if (node >= nNodes) return;
  int beg = rowptr[node];
  int end = rowptr[node + 1];

  float ax = 0.f, ay = 0.f, az = 0.f, aw = 0.f;
  // Chunk edge list by 32: one coalesced index load, broadcast via shuffles.
  for (int cbeg = beg; cbeg < end; cbeg += 32) {
    int cl = end - cbeg;
    if (cl > 32) cl = 32;
    int myIdx = (cbeg + lane < end) ? srcIdx[cbeg + lane] : 0;
    for (int j = 0; j < cl; ++j) {
      int s = __shfl(myIdx, j, 32);
      float4 v = *((const float4*)(h + (size_t)s * FEAT) + lane);
      ax += v.x; ay += v.y; az += v.z; aw += v.w;
    }
  }

  float idg = 1.0f / fmaxf((float)(end - beg), 1.0f);
  float epsv = 1.0f + epsArr[layer];
  float4 hv = *((const float4*)(h + (size_t)node * FEAT) + lane);
  float4 zv;
  zv.x = epsv * hv.x + ax * idg;
  zv.y = epsv * hv.y + ay * idg;
  zv.z = epsv * hv.z + az * idg;
  zv.w = epsv * hv.w + aw * idg;
  *((float4*)(z + (size_t)node * FEAT) + lane) = zv;
}

// GEMM out = relu(z @ W + b). Block = 256 threads = 8 wave32s, owns 16 rows.
// Wave w computes the 16x16 tile at columns [16w, 16w+16) via f32 WMMA over K=128.
template <int NOUT, bool RELU>
__global__ __launch_bounds__(256) void gin_gemm_kernel(const float* __restrict__ z,
                                                       const float* __restrict__ W,
                                                       const float* __restrict__ bias,
                                                       int nNodes, float* __restrict__ out) {
  __shared__ float zs[16][FEAT + ZPAD];
  const int mbase = blockIdx.x * 16;

#ifdef USE_ASYNC_LDS
  // Async path: stream the 16x128 z tile straight into LDS (B128 chunks, no VGPR staging).
  for (int i = threadIdx.x; i < 16 * 32; i += 256) {
    int r = i >> 5;       // row 0..15
    int c4 = i & 31;      // float4 column 0..31
    int m = mbase + r;
    if (m >= nNodes) m = nNodes - 1;
    const float* gp = z + (size_t)m * FEAT + c4 * 4;
    __builtin_amdgcn_global_load_async_to_lds_b128((gv4i_p)gp, (lv4i_p)&zs[r][c4 * 4], 0, 0);
  }
  asm volatile("s_wait_asynccnt 0x0" ::: "memory");
#else
  for (int i = threadIdx.x; i < 16 * FEAT; i += 256) {
    int r = i >> 7;
    int c = i & (FEAT - 1);
    int m = mbase + r;
    if (m >= nNodes) m = nNodes - 1;
    zs[r][c] = z[(size_t)m * FEAT + c];
  }
#endif
  __syncthreads();

  const int wave = threadIdx.x >> 5;
  const int lane = threadIdx.x & 31;
  const int hi = lane >> 4;  // half-wave select
  const int lo = lane & 15;  // A: row M; B/C: column N within tile

  if (wave * 16 < NOUT) {
    const int nbase = wave * 16;
    v8f acc = {0.f, 0.f, 0.f, 0.f, 0.f, 0.f, 0.f, 0.f};

    // A 16x4: lanes 0-15 -> (M=lane, K=k0+0/1), lanes 16-31 -> (M=lane-16, K=k0+2/3)
    // B 4x16: lanes 0-15 -> (K=k0+0/1, N=lane), lanes 16-31 -> (K=k0+2/3, N=lane-16)
#pragma unroll
    for (int k0 = 0; k0 < FEAT; k0 += 4) {
      const int ka = k0 + 2 * hi;
      v2f a, b;
      a.x = zs[lo][ka];
      a.y = zs[lo][ka + 1];
      b.x = W[(size_t)ka * NOUT + nbase + lo];
      b.y = W[(size_t)(ka + 1) * NOUT + nbase + lo];
      acc = __builtin_amdgcn_wmma_f32_16x16x4_f32(false, a, false, b, (short)0, acc,
                                                  false, false);
    }

    // C/D layout: VGPR i -> M = mbase + 8*hi + i, N = nbase + lo.
    const int n = nbase + lo;
    const float bn = bias[n];
#pragma unroll
    for (int i = 0; i < 8; ++i) {
      int m = mbase + hi * 8 + i;
      if (m < nNodes) {
        float v = acc[i] + bn;
        if (RELU) v = fmaxf(v, 0.0f);
        out[(size_t)m * NOUT + n] = v;
      }
    }
  }
}

extern "C" void kernel_launch(void* const* d_in, const int* in_sizes, int n_in,
                              void* d_out, int out_size, void* d_ws, size_t ws_size,
                              hipStream_t stream) {
  const float* features = (const float*)d_in[0];
  const int*   src      = (const int*)d_in[1];
  const int*   dst      = (const int*)d_in[2];
  const float* W0 = (const float*)d_in[3];
  const float* b0 = (const float*)d_in[4];
  const float* W1 = (const float*)d_in[5];
  const float* b1 = (const float*)d_in[6];
  const float* W2 = (const float*)d_in[7];
  const float* b2 = (const float*)d_in[8];
  const float* W3 = (const float*)d_in[9];
  const float* b3 = (const float*)d_in[10];
  const float* eps = (const float*)d_in[11];

  const int nNodes = in_sizes[0] / FEAT;  // 100000
  const int nEdges = in_sizes[1];         // 1600000

  // Workspace: zbuf[n*128] f32 | hbuf[n*128] f32 | rowptr[n+1] | fill[n] | srcIdx[nEdges]
  float* zbuf = (float*)d_ws;
  float* hbuf = zbuf + (size_t)nNodes * FEAT;
  int* rowptr = (int*)(hbuf + (size_t)nNodes * FEAT);
  int* fill   = rowptr + (nNodes + 1);
  int* srcIdx = fill + nNodes;

  const int nodeBlks    = (nNodes + 255) / 256;
  const int edgeBlks    = (nEdges + 255) / 256;
  const int nodeWaveBlk = (nNodes * 32 + 255) / 256;  // wave-per-node
  const int gemmBlks    = (nNodes + 15) / 16;

  // ---- CSR build (dst -> src list) ----
  gin_zero_i32<<<nodeBlks, 256, 0, stream>>>(fill, nNodes);
  gin_count_kernel<<<edgeBlks, 256, 0, stream>>>(dst, fill, nEdges);
  gin_scan_kernel<<<1, 1024, 0, stream>>>(fill, rowptr, nNodes);
  gin_zero_i32<<<nodeBlks, 256, 0, stream>>>(fill, nNodes);
  gin_scatter_kernel<<<edgeBlks, 256, 0, stream>>>(src, dst, rowptr, fill, srcIdx, nEdges);

  // ---- layer 0: features -> hbuf ----
  gin_aggz_kernel<<<nodeWaveBlk, 256, 0, stream>>>(features, rowptr, srcIdx, eps, 0, nNodes, zbuf);
  gin_gemm_kernel<128, true><<<gemmBlks, 256, 0, stream>>>(zbuf, W0, b0, nNodes, hbuf);

  // ---- layer 1: hbuf -> hbuf ----
  gin_aggz_kernel<<<nodeWaveBlk, 256, 0, stream>>>(hbuf, rowptr, srcIdx, eps, 1, nNodes, zbuf);
  gin_gemm_kernel<128, true><<<gemmBlks, 256, 0, stream>>>(zbuf, W1, b1, nNodes, hbuf);

  // ---- layer 2: hbuf -> hbuf ----
  gin_aggz_kernel<<<nodeWaveBlk, 256, 0, stream>>>(hbuf, rowptr, srcIdx, eps, 2, nNodes, zbuf);
  gin_gemm_kernel<128, true><<<gemmBlks, 256, 0, stream>>>(zbuf, W2, b2, nNodes, hbuf);

  // ---- layer 3: hbuf -> d_out (N=16, no ReLU) ----
  gin_aggz_kernel<<<nodeWaveBlk, 256, 0, stream>>>(hbuf, rowptr, srcIdx, eps, 3, nNodes, zbuf);
  gin_gemm_kernel<16, false><<<gemmBlks, 256, 0, stream>>>(zbuf, W3, b3, nNodes, (float*)d_out);
}